// ConvolutionalCapsule_51754355917253
// MI455X (gfx1250) — compile-verified
//
#include <hip/hip_runtime.h>
#include <cmath>

// ---------------- problem constants (from the reference) ----------------
#define Bb    8
#define Hh    16
#define CIc   16
#define Oo    16
#define Kk    3
#define Pp    14          // (H-K)/S + 1
#define PP2   196         // P*P
#define K2CI  144         // K*K*CI
#define EPSf  1e-9f

#define NBP   (Bb*PP2)    // 1568  (= "n" dimension of pose_unroll)
#define NROW  (NBP*4)     // 6272  GEMM M (rows = (n,p))
#define MT    (NROW/16)   // 392   M tiles of 16

static const size_t NVOTES = (size_t)NBP * K2CI * Oo * 16; // 57,802,752 f32
static const size_t NRR    = (size_t)NBP * K2CI * Oo;      //  3,612,672 f32
static const size_t NMEAN  = (size_t)NBP * Oo * 16;        //    401,408 f32
static const size_t NACT   = (size_t)NBP * Oo;             //     25,088 f32

typedef __attribute__((ext_vector_type(2))) float v2f;
typedef __attribute__((ext_vector_type(8))) float v8f;

// =======================================================================
// Kernel 1: votes[n,k,o,p,r] = sum_q pose_patch[n,k,p,q] * w[k,o,q,r]
// Per fixed k this is GEMM  M=6272 (rows=(n,p)), K=4 (q), N=64 (cols=(o,r))
// -> V_WMMA_F32_16X16X4_F32, one wave per (M-tile, k), 4 N-tiles unrolled.
// A 16x4 f32 layout: lanes 0-15: M=lane, VGPR0->K=0, VGPR1->K=1;
//                    lanes 16-31: M=lane-16, VGPR0->K=2, VGPR1->K=3.
// B 4x16 f32 layout (mirror): lanes 0-15 hold K=0,1; lanes 16-31 hold K=2,3;
//                    N = lane&15.
// C/D 16x16 f32: VGPR v, lane l -> M = v + 8*(l>>4), N = l&15.
// =======================================================================
__global__ __launch_bounds__(32)
void votes_wmma_kernel(const float* __restrict__ pose,   // [B,H,H,CI,16]
                       const float* __restrict__ w,      // [K2CI,O,4,4]
                       float* __restrict__ votes)        // [NBP,K2CI,O,16]
{
    const int tile_m = blockIdx.x;          // 0..MT-1
    const int k      = blockIdx.y;          // 0..K2CI-1
    const int lane   = threadIdx.x;         // 0..31
    const int half   = lane >> 4;
    const int l15    = lane & 15;

    const int ci  = k % CIc;
    const int kk2 = k / CIc;
    const int kh  = kk2 / Kk;
    const int kw  = kk2 % Kk;
    const int q0  = 2 * half;               // K values held by this lane half

    // ---- A fragment straight from implicit im2col of inputs_pose
    const int rowA = tile_m * 16 + l15;     // global row = n*4 + p
    const int nA   = rowA >> 2;
    const int pA   = rowA & 3;
    const int bA   = nA / PP2;
    const int p2A  = nA % PP2;
    const int phA  = p2A / Pp, pwA = p2A % Pp;
    const size_t poseBase =
        ((((size_t)bA * Hh + (phA + kh)) * Hh + (pwA + kw)) * CIc + ci) * 16
        + (size_t)pA * 4;
    v2f a;
    a.x = pose[poseBase + q0];
    a.y = pose[poseBase + q0 + 1];

    #pragma unroll
    for (int tn = 0; tn < 4; ++tn) {
        const int col = tn * 16 + l15;      // global col = o*4 + r
        const int o   = col >> 2;
        const int r   = col & 3;
        // w[k][o][q][r] flat = (k*O+o)*16 + q*4 + r
        const size_t wBase = ((size_t)k * Oo + o) * 16 + r;
        v2f bf;
        bf.x = w[wBase + (size_t)q0 * 4];
        bf.y = w[wBase + (size_t)(q0 + 1) * 4];

        v8f c = {};
        v8f d = __builtin_amdgcn_wmma_f32_16x16x4_f32(
                    /*neg_a=*/false, a, /*neg_b=*/false, bf,
                    /*c_mod=*/(short)0, c, /*reuse_a=*/false, /*reuse_b=*/false);

        #pragma unroll
        for (int v = 0; v < 8; ++v) {
            const int rowD = tile_m * 16 + v + 8 * half;
            const int nD   = rowD >> 2;
            const int pD   = rowD & 3;
            votes[(((size_t)nD * K2CI + k) * Oo + o) * 16 + (size_t)pD * 4 + r] = d[v];
        }
    }
}

// =======================================================================
// Kernel 2: rr init: 1/(O * #parents covering child pixel). S=1 so the
// per-axis cover count is closed-form; no host-side counts table needed.
// =======================================================================
__device__ __forceinline__ int cov_count(int h) {
    int lo = h - (Kk - 1); if (lo < 0) lo = 0;
    int hi = h;            if (hi > Pp - 1) hi = Pp - 1;
    return hi - lo + 1;
}

__global__ void rr_init_kernel(float* __restrict__ rr) {
    size_t idx = (size_t)blockIdx.x * blockDim.x + threadIdx.x;
    if (idx >= NRR) return;
    const int kq  = (int)((idx / Oo) % K2CI);
    const int p2  = (int)((idx / ((size_t)Oo * K2CI)) % PP2);
    const int kk2 = kq / CIc;
    const int kh  = kk2 / Kk, kw = kk2 % Kk;
    const int ph  = p2 / Pp,  pw = p2 % Pp;
    const int cnt = cov_count(ph + kh) * cov_count(pw + kw);
    rr[idx] = 1.0f / ((float)Oo * (float)cnt);
}

// =======================================================================
// Kernel 3 (M-step): one wave per (b,p2,o). Lanes split k=0..143,
// accumulate S0 = sum rp, S1[d] = sum rp*v, S2[d] = sum rp*v^2,
// wave32 shuffle all-reduce, lane 0 emits mean / 1/var / sum log(2*pi*var)
// / act_j.  var = (S2 - 2 m S1 + m^2 S0)/rsum + EPS with rsum = S0 + EPS
// (exactly the reference's EPS placement).
// =======================================================================
__global__ __launch_bounds__(32)
void mstep_kernel(const float* __restrict__ votes, const float* __restrict__ rr,
                  const float* __restrict__ act,                    // [B,H,H,CI]
                  const float* __restrict__ beta_a, const float* __restrict__ beta_v,
                  float* __restrict__ mean, float* __restrict__ invvar,
                  float* __restrict__ lvsum, float* __restrict__ actj,
                  float inv_temp)
{
    const int idx = blockIdx.x;          // 0 .. NBP*O-1
    const int o   = idx % Oo;
    const int bp  = idx / Oo;            // b*P2 + p2
    const int b   = bp / PP2;
    const int p2  = bp % PP2;
    const int ph  = p2 / Pp, pw = p2 % Pp;
    const int lane = threadIdx.x;

    float s0 = 0.f;
    float s1[16], s2[16];
    #pragma unroll
    for (int d = 0; d < 16; ++d) { s1[d] = 0.f; s2[d] = 0.f; }

    for (int kq = lane; kq < K2CI; kq += 32) {
        const int ci  = kq % CIc;
        const int kk2 = kq / CIc;
        const int kh  = kk2 / Kk, kw = kk2 % Kk;
        const float r = rr[((size_t)bp * K2CI + kq) * Oo + o];
        const float a = act[(((size_t)b * Hh + (ph + kh)) * Hh + (pw + kw)) * CIc + ci];
        const float rp = r * a;
        s0 += rp;
        const float4* vp =
            (const float4*)(votes + (((size_t)bp * K2CI + kq) * Oo + o) * 16);
        #pragma unroll
        for (int j = 0; j < 4; ++j) {
            const float4 v = vp[j];
            s1[4*j+0] += rp * v.x;  s2[4*j+0] += rp * v.x * v.x;
            s1[4*j+1] += rp * v.y;  s2[4*j+1] += rp * v.y * v.y;
            s1[4*j+2] += rp * v.z;  s2[4*j+2] += rp * v.z * v.z;
            s1[4*j+3] += rp * v.w;  s2[4*j+3] += rp * v.w * v.w;
        }
    }

    // wave32 all-reduce
    #pragma unroll
    for (int off = 16; off >= 1; off >>= 1) {
        s0 += __shfl_xor(s0, off, 32);
        #pragma unroll
        for (int d = 0; d < 16; ++d) {
            s1[d] += __shfl_xor(s1[d], off, 32);
            s2[d] += __shfl_xor(s2[d], off, 32);
        }
    }

    if (lane == 0) {
        const float rsum = s0 + EPSf;
        const float bv = beta_v[o];
        float cost = 0.f, slv = 0.f;
        #pragma unroll
        for (int d = 0; d < 16; ++d) {
            const float m   = s1[d] / rsum;
            const float var = (s2[d] - 2.f * m * s1[d] + m * m * s0) / rsum + EPSf;
            const float lv  = logf(var);
            cost += (bv + 0.5f * lv) * rsum;
            slv  += lv;
            mean  [(size_t)idx * 16 + d] = m;
            invvar[(size_t)idx * 16 + d] = 1.0f / var;
        }
        lvsum[idx] = slv + 16.0f * logf(6.283185307179586f);  // + sum_d log(2*pi)
        actj[idx]  = 1.0f / (1.0f + expf(-inv_temp * (beta_a[o] - cost)));
    }
}

// =======================================================================
// Kernel 4 (E-step part 1): zz[b,p2,k,o] = log(act_j+EPS)
//   - 0.5 * ( sum_d (v-m)^2 / var  +  sum_d log(2*pi*var) )
// =======================================================================
__global__ void estep_zz_kernel(const float* __restrict__ votes,
                                const float* __restrict__ mean,
                                const float* __restrict__ invvar,
                                const float* __restrict__ lvsum,
                                const float* __restrict__ actj,
                                float* __restrict__ zz)
{
    size_t idx = (size_t)blockIdx.x * blockDim.x + threadIdx.x;
    if (idx >= NRR) return;
    const int    o  = (int)(idx % Oo);
    const size_t t  = idx / Oo;            // bp*K2CI + kq
    const size_t bp = t / K2CI;
    const size_t mi = (bp * Oo + o) * 16;

    const float4* vp = (const float4*)(votes + idx * 16);
    const float4* mp = (const float4*)(mean + mi);
    const float4* ip = (const float4*)(invvar + mi);
    float acc = 0.f;
    #pragma unroll
    for (int j = 0; j < 4; ++j) {
        const float4 v = vp[j], m = mp[j], iv = ip[j];
        float dx;
        dx = v.x - m.x; acc += dx * dx * iv.x;
        dx = v.y - m.y; acc += dx * dx * iv.y;
        dx = v.z - m.z; acc += dx * dx * iv.z;
        dx = v.w - m.w; acc += dx * dx * iv.w;
    }
    const float logp = -0.5f * (acc + lvsum[bp * Oo + o]);
    zz[idx] = logf(actj[bp * Oo + o] + EPSf) + logp;
}

// =======================================================================
// Kernel 5 (E-step part 2): per-child softmax. Inverts the reference's
// segment scatter analytically: child pixel (h,w) is voted for by parents
// ph in [max(0,h-K+1), min(P-1,h)] (same for w), kernel offset kh = h-ph.
// For each (b,h,w,ci): joint softmax over all (parent, o) entries,
// rr_new = exp(z-g)/(segsum + EPS), written back into rr.
// =======================================================================
__global__ void child_softmax_kernel(const float* __restrict__ zz,
                                     float* __restrict__ rr)
{
    int idx = blockIdx.x * blockDim.x + threadIdx.x;
    if (idx >= Bb * Hh * Hh * CIc) return;
    const int ci = idx % CIc;
    int t = idx / CIc;
    const int wcol = t % Hh;  t /= Hh;
    const int hrow = t % Hh;
    const int b    = t / Hh;

    int phlo = hrow - (Kk - 1); if (phlo < 0) phlo = 0;
    int phhi = hrow;            if (phhi > Pp - 1) phhi = Pp - 1;
    int pwlo = wcol - (Kk - 1); if (pwlo < 0) pwlo = 0;
    int pwhi = wcol;            if (pwhi > Pp - 1) pwhi = Pp - 1;

    // pass 1: segment max
    float gmax = -INFINITY;
    for (int ph = phlo; ph <= phhi; ++ph)
        for (int pw = pwlo; pw <= pwhi; ++pw) {
            const int kh = hrow - ph, kw = wcol - pw;
            const int p2 = ph * Pp + pw;
            const int kq = (kh * Kk + kw) * CIc + ci;
            const size_t base = (((size_t)b * PP2 + p2) * K2CI + kq) * Oo;
            for (int o = 0; o < Oo; ++o) gmax = fmaxf(gmax, zz[base + o]);
        }
    // pass 2: segment expsum
    float denom = 0.f;
    for (int ph = phlo; ph <= phhi; ++ph)
        for (int pw = pwlo; pw <= pwhi; ++pw) {
            const int kh = hrow - ph, kw = wcol - pw;
            const int p2 = ph * Pp + pw;
            const int kq = (kh * Kk + kw) * CIc + ci;
            const size_t base = (((size_t)b * PP2 + p2) * K2CI + kq) * Oo;
            for (int o = 0; o < Oo; ++o) denom += expf(zz[base + o] - gmax);
        }
    const float inv = 1.0f / (denom + EPSf);
    // pass 3: write normalized rr
    for (int ph = phlo; ph <= phhi; ++ph)
        for (int pw = pwlo; pw <= pwhi; ++pw) {
            const int kh = hrow - ph, kw = wcol - pw;
            const int p2 = ph * Pp + pw;
            const int kq = (kh * Kk + kw) * CIc + ci;
            const size_t base = (((size_t)b * PP2 + p2) * K2CI + kq) * Oo;
            for (int o = 0; o < Oo; ++o)
                rr[base + o] = expf(zz[base + o] - gmax) * inv;
        }
}

// =======================================================================
// Kernel 6: pack (pose_out, act_out) into d_out (concatenated flat).
// mean is already laid out as [B,P2,O,16] == (B,P,P,O,16) flat.
// =======================================================================
__global__ void finalize_kernel(const float* __restrict__ mean,
                                const float* __restrict__ actj,
                                float* __restrict__ out)
{
    size_t idx = (size_t)blockIdx.x * blockDim.x + threadIdx.x;
    if (idx < NMEAN)            out[idx] = mean[idx];
    else if (idx < NMEAN + NACT) out[idx] = actj[idx - NMEAN];
}

// =======================================================================
extern "C" void kernel_launch(void* const* d_in, const int* in_sizes, int n_in,
                              void* d_out, int out_size, void* d_ws, size_t ws_size,
                              hipStream_t stream)
{
    (void)in_sizes; (void)n_in; (void)out_size; (void)ws_size;
    const float* pose   = (const float*)d_in[0];  // [B,H,H,CI,16]
    const float* act    = (const float*)d_in[1];  // [B,H,H,CI,1]
    const float* w      = (const float*)d_in[2];  // [1,K2CI,O,4,4]
    const float* beta_a = (const float*)d_in[3];  // [O]
    const float* beta_v = (const float*)d_in[4];  // [O]
    float* out = (float*)d_out;

    // workspace carve-up (floats); all sections 64B aligned (multiples of 16)
    float* ws    = (float*)d_ws;
    size_t off   = 0;
    float* votes = ws + off; off += NVOTES;
    float* rr    = ws + off; off += NRR;
    float* zz    = ws + off; off += NRR;
    float* mean  = ws + off; off += NMEAN;
    float* ivar  = ws + off; off += NMEAN;
    float* lvs   = ws + off; off += NACT;
    float* aj    = ws + off; off += NACT;

    // 1) votes via WMMA f32 16x16x4
    votes_wmma_kernel<<<dim3(MT, K2CI), dim3(32), 0, stream>>>(pose, w, votes);

    // 2) rr init
    {
        const int thr = 256;
        const int blk = (int)((NRR + thr - 1) / thr);
        rr_init_kernel<<<blk, thr, 0, stream>>>(rr);
    }

    // 3) EM routing, 3 iterations
    for (int it = 0; it < 3; ++it) {
        const float inv_temp = 0.01f * (1.0f - powf(0.95f, (float)(it + 1)));
        mstep_kernel<<<NBP * Oo, 32, 0, stream>>>(votes, rr, act, beta_a, beta_v,
                                                  mean, ivar, lvs, aj, inv_temp);
        if (it < 2) {
            const int thr = 256;
            estep_zz_kernel<<<(int)((NRR + thr - 1) / thr), thr, 0, stream>>>(
                votes, mean, ivar, lvs, aj, zz);
            child_softmax_kernel<<<(Bb * Hh * Hh * CIc + thr - 1) / thr, thr, 0,
                                   stream>>>(zz, rr);
        }
    }

    // 4) outputs
    {
        const size_t ntot = NMEAN + NACT;
        const int thr = 256;
        finalize_kernel<<<(int)((ntot + thr - 1) / thr), thr, 0, stream>>>(mean, aj, out);
    }
}